// VolumeRenderer_17514876634254
// MI455X (gfx1250) — compile-verified
//
#include <hip/hip_runtime.h>
#include <hip/hip_bf16.h>
#include <math.h>

// ---------------------------------------------------------------------------
// BlockNeRF-style volume renderer for MI455X (gfx1250, wave32).
//
// Design:
//  * one ray per lane; 1024 blocks x 128 threads covers all 131072 rays
//    exactly -> EXEC always all-ones (WMMA-legal, no divergence at launch).
//  * affine factorization: logit_c(t) = oW[c] + t*dW[c]. The two [N,3]x[3,4]
//    projections (origins and directions vs [Wd | Wc]) are computed with
//    V_WMMA_F32_16X16X4_F32: A = W^T (channels x coords), B = 16 ray vectors.
//    Half-wave SWAPX16 ds_swizzles marshal the documented VGPR layouts.
//  * coarse pass: serial transmittance per lane, unnormalized CDF stored in
//    LDS (stride 65 -> conflict-free), no barriers (private slices).
//  * fine pass: u = linspace is increasing and the CDF is monotone, so the
//    inverse-CDF samples are generated in sorted order; a streaming 3-way
//    merge replaces jnp.sort + searchsorted entirely (O(S) per ray).
// ---------------------------------------------------------------------------

#define N_RAYS   131072
#define NSAMP    64
#define NIMP     64
#define NEAR_T   2.0f
#define FAR_T    6.0f
#define EPS_Wc   1e-5f
#define EPS_Tc   1e-10f
#define FARDIST  1e10f
#define BLOCK    128

typedef __attribute__((ext_vector_type(2))) float v2f;
typedef __attribute__((ext_vector_type(8))) float v8f;

// half-wave swap: lane L <-> lane L^16 (ds_swizzle group-of-32, xor=0x10)
__device__ __forceinline__ float swapx16(float x) {
  return __int_as_float(__builtin_amdgcn_ds_swizzle(__float_as_int(x), 0x401F));
}

__device__ __forceinline__ v8f wmma4(v2f a, v2f b) {
  v8f c = {};
  // (neg_a, A, neg_b, B, c_mod, C, reuse_a, reuse_b)
  return __builtin_amdgcn_wmma_f32_16x16x4_f32(false, a, false, b, (short)0, c,
                                               false, false);
}

__device__ __forceinline__ float softplus_f(float x) {
  // log1p(exp(x)), overflow-safe
  return fmaxf(x, 0.0f) + __logf(1.0f + __expf(-fabsf(x)));
}
__device__ __forceinline__ float sigmoid_f(float x) {
  return 1.0f / (1.0f + __expf(-x));
}

// project vec (x,y,z) per-lane against W^T held in A; 2 WMMAs per wave cover
// all 32 rays (16 columns each), SWAPX16 merges results back to owner lanes.
__device__ __forceinline__ void proj4(float x, float y, float z, v2f A,
                                      bool lo, float out[4]) {
  // WMMA #1: columns = rays in lanes 0..15
  //   B vgpr0: lanes<16 -> K0 (x of own ray); lanes>=16 -> K2 (z of lane-16)
  //   B vgpr1: lanes<16 -> K1 (y);            lanes>=16 -> K3 pad (0)
  float zs = swapx16(z);
  v2f B1;
  B1.x = lo ? x : zs;
  B1.y = lo ? y : 0.0f;
  v8f D1 = wmma4(A, B1);

  // WMMA #2: columns = rays in lanes 16..31
  float xs = swapx16(x);
  float ys = swapx16(y);
  v2f B2;
  B2.x = lo ? xs : z;
  B2.y = lo ? ys : 0.0f;
  v8f D2 = wmma4(A, B2);

  // D vgpr c, lanes 0..15 = channel c for the 16 columns; merge halves.
#pragma unroll
  for (int c = 0; c < 4; ++c) {
    float hi = swapx16(D2[c]);
    out[c] = lo ? D1[c] : hi;
  }
}

__global__ __launch_bounds__(BLOCK) void nerf_render_kernel(
    const float* __restrict__ ro, const float* __restrict__ rd,
    const float* __restrict__ Wd, const float* __restrict__ Wc,
    float* __restrict__ out) {
  __shared__ float s_cw[BLOCK * 65];  // stride-65 padding: conflict-free

  const int tid  = blockIdx.x * BLOCK + threadIdx.x;  // ray id (exact cover)
  const int lane = threadIdx.x & 31;
  const bool lo  = lane < 16;

  // ---- build A = W^T in 16x4 f32 A-layout (vgpr0: K0|K2, vgpr1: K1|K3) ----
  // channel 0 = Wd column, channels 1..3 = Wc columns; rows 4..15 are zero.
  const int ch = lane & 15;
  float a0 = 0.0f, a1 = 0.0f;
  if (ch < 4) {
    const int k0 = lo ? 0 : 2;
    a0 = (ch == 0) ? Wd[k0] : Wc[k0 * 3 + (ch - 1)];
    if (lo) a1 = (ch == 0) ? Wd[1] : Wc[3 + (ch - 1)];  // K=1; K=3 pad = 0
  }
  v2f A;
  A.x = a0;
  A.y = a1;

  const float ox = ro[tid * 3 + 0], oy = ro[tid * 3 + 1], oz = ro[tid * 3 + 2];
  const float dx = rd[tid * 3 + 0], dy = rd[tid * 3 + 1], dz = rd[tid * 3 + 2];

  float oW[4], dW[4];
  proj4(ox, oy, oz, A, lo, oW);   // logit_c(t) = oW[c] + t * dW[c]
  proj4(dx, dy, dz, A, lo, dW);

  // ----------------------------- coarse pass ------------------------------
  float* cw = &s_cw[threadIdx.x * 65];
  const float step = (FAR_T - NEAR_T) / (float)(NSAMP - 1);

  float T = 1.0f, r = 0.0f, g = 0.0f, b = 0.0f;
  float depth = 0.0f, acc = 0.0f, csum = 0.0f;

#pragma unroll 4
  for (int s = 0; s < NSAMP; ++s) {
    const float t     = NEAR_T + step * (float)s;
    const float sig   = softplus_f(oW[0] + t * dW[0]);
    const float dist  = (s < NSAMP - 1) ? step : FARDIST;
    const float alpha = 1.0f - __expf(-sig * dist);
    const float w     = alpha * T;
    T *= (1.0f - alpha + EPS_Tc);
    r += w * sigmoid_f(oW[1] + t * dW[1]);
    g += w * sigmoid_f(oW[2] + t * dW[2]);
    b += w * sigmoid_f(oW[3] + t * dW[3]);
    depth += w * t;
    acc += w;
    csum += w + EPS_Wc;   // unnormalized CDF (pdf adds EPS_W per bin)
    cw[s] = csum;
  }

  out[tid * 3 + 0]        = r + (1.0f - acc);
  out[tid * 3 + 1]        = g + (1.0f - acc);
  out[tid * 3 + 2]        = b + (1.0f - acc);
  out[3 * N_RAYS + tid]   = depth;
  out[4 * N_RAYS + tid]   = acc;

  // ------------------- fine pass: streaming 3-way merge -------------------
  const float total     = csum;
  const float inv_total = 1.0f / total;
  const float ustep     = 1.0f / (float)(NIMP - 1);

  int ai = 0, ii = 0, jj = 0;
  float ta = NEAR_T;

  // lazily generate importance sample for current ii (u increasing -> jj
  // advances monotonically; total extra work O(NSAMP) per ray)
  auto nextB = [&]() -> float {
    const float u      = (float)ii * ustep;
    const float target = u * total;
    while (jj < NSAMP && cw[jj] <= target) ++jj;           // searchsorted-right
    const float c0 = (jj == 0) ? 0.0f : cw[jj - 1] * inv_total;
    const float c1 = ((jj < NSAMP) ? cw[jj] : cw[NSAMP - 1]) * inv_total;
    float den = c1 - c0;
    if (den < EPS_Wc) den = 1.0f;
    const float frac = (u - c0) / den;
    const int b0 = (jj < NSAMP - 1) ? jj : (NSAMP - 1);
    const int b1 = (jj + 1 < NSAMP - 1) ? (jj + 1) : (NSAMP - 1);
    const float bin0 = NEAR_T + step * (float)b0;
    const float bin1 = NEAR_T + step * (float)b1;
    return bin0 + frac * (bin1 - bin0);
  };

  float tb = nextB();

  auto popMin = [&]() -> float {
    float t;
    if (ai < NSAMP && (ii >= NIMP || ta <= tb)) {
      t = ta;
      ++ai;
      ta = NEAR_T + step * (float)ai;
    } else {
      t = tb;
      ++ii;
      tb = (ii < NIMP) ? nextB() : (2.0f * FARDIST);
    }
    return t;
  };

  float fT = 1.0f, fr = 0.0f, fg = 0.0f, fb = 0.0f, fdepth = 0.0f, facc = 0.0f;

  auto shade = [&](float t, float dist) {
    const float sig   = softplus_f(oW[0] + t * dW[0]);
    const float alpha = 1.0f - __expf(-sig * dist);
    const float w     = alpha * fT;
    fT *= (1.0f - alpha + EPS_Tc);
    fr += w * sigmoid_f(oW[1] + t * dW[1]);
    fg += w * sigmoid_f(oW[2] + t * dW[2]);
    fb += w * sigmoid_f(oW[3] + t * dW[3]);
    fdepth += w * t;
    facc += w;
  };

  float prev = popMin();
  for (int k = 1; k < NSAMP + NIMP; ++k) {
    const float t = popMin();
    shade(prev, t - prev);
    prev = t;
  }
  shade(prev, FARDIST);  // last merged sample: dist = FAR_DIST

  out[5 * N_RAYS + tid * 3 + 0] = fr + (1.0f - facc);
  out[5 * N_RAYS + tid * 3 + 1] = fg + (1.0f - facc);
  out[5 * N_RAYS + tid * 3 + 2] = fb + (1.0f - facc);
  out[8 * N_RAYS + tid]         = fdepth;
  out[9 * N_RAYS + tid]         = facc;
}

extern "C" void kernel_launch(void* const* d_in, const int* in_sizes, int n_in,
                              void* d_out, int out_size, void* d_ws,
                              size_t ws_size, hipStream_t stream) {
  (void)in_sizes; (void)n_in; (void)out_size; (void)d_ws; (void)ws_size;
  const float* ro = (const float*)d_in[0];   // [N,3]
  const float* rd = (const float*)d_in[1];   // [N,3]
  const float* Wd = (const float*)d_in[3];   // [3,1]
  const float* Wc = (const float*)d_in[4];   // [3,3]
  float* out = (float*)d_out;                // 10*N floats

  dim3 grid(N_RAYS / BLOCK), block(BLOCK);
  nerf_render_kernel<<<grid, block, 0, stream>>>(ro, rd, Wd, Wc, out);
}